// DiffusionBlock_27264452395008
// MI455X (gfx1250) — compile-verified
//
#include <hip/hip_runtime.h>
#include <hip/hip_bf16.h>
#include <stdint.h>

// Diffusion step: out = u + tau*div(D*grad u), fused 9-point stencil.
// Memory-bound (AI ~1.2 flop/byte, ~330 MB unique HBM traffic -> ~14us floor).
// All bulk data movement goes through the CDNA5 Tensor Data Mover:
//   wave0: u tile (18x256 -> LDS pitch 258 via TDM pad feature)
//   wave1/2/3: a/b/c tiles (17x258, never need clamping -> plain 2D DMA)
// Compute reads exclusively from LDS; only scalars + halo cols use plain loads.

#define TW 256              // tile width  == blockDim.x (8 wave32 waves)
#define TH 16               // tile height (rows per block)
#define HH 1024
#define WW 1024
#define WPAD 1026           // padded pitch of a/b/c
#define PH (TH + 1)         // a/b/c tile rows   (17)
#define PW (TW + 2)         // a/b/c tile cols   (258) == LDS pitch

typedef unsigned int uint32x4 __attribute__((ext_vector_type(4)));
typedef int          int32x4  __attribute__((ext_vector_type(4)));
typedef int          int32x8  __attribute__((ext_vector_type(8)));

#if __has_builtin(__builtin_amdgcn_tensor_load_to_lds) && __has_builtin(__builtin_amdgcn_s_wait_tensorcnt)
#define HAVE_TDM 1
#else
#define HAVE_TDM 0
#endif

#if HAVE_TDM
#if __has_include(<hip/amd_detail/amd_gfx1250_TDM.h>)
// therock-10.0 headers => 6-arg builtin
#define TDM_CALL(g0, g1, g2, g3) \
    __builtin_amdgcn_tensor_load_to_lds((g0), (g1), (g2), (g3), (int32x8){0,0,0,0,0,0,0,0}, 0)
#else
// ROCm 7.2 clang-22 => 5-arg builtin
#define TDM_CALL(g0, g1, g2, g3) \
    __builtin_amdgcn_tensor_load_to_lds((g0), (g1), (g2), (g3), 0)
#endif

// DMA an (nrows x ncols) f32 tile (global row stride gstride elements) into LDS
// at byte offset lds_byte. padbits!=0 inserts LDS padding (for u: 2 dwords after
// every 256 -> pitch 258); padbits==0 packs rows contiguously (pitch == ncols).
__device__ __forceinline__ void tdm_load_2d(const float* gsrc, unsigned lds_byte,
                                            int nrows, int ncols, int gstride,
                                            unsigned padbits)
{
    uint64_t ga = (uint64_t)(uintptr_t)gsrc;
    uint32x4 g0;
    g0[0] = 1u;                                        // count=1 (valid descriptor)
    g0[1] = lds_byte;                                  // lds_addr (bytes)
    g0[2] = (unsigned)(ga & 0xFFFFFFFFu);              // global_addr[31:0]
    g0[3] = (unsigned)((ga >> 32) & 0x01FFFFFFu)       // global_addr[56:32]
          | (2u << 30);                                // type=2 ("image")
    int32x8 g1;
    g1[0] = (int)((2u << 16) | padbits);               // data_size=4B | pad cfg
    g1[1] = (int)((unsigned)(ncols & 0xFFFF) << 16);   // tensor_dim0[15:0] @63:48
    g1[2] = (int)((unsigned)nrows << 16);              // dim0 hi(0) | tensor_dim1 lo
    g1[3] = (int)((unsigned)ncols << 16);              // dim1 hi(0) | tile_dim0
    g1[4] = nrows;                                     // tile_dim1 | tile_dim2=0
    g1[5] = gstride;                                   // tensor_dim0_stride[31:0]
    g1[6] = 0;                                         // stride0 hi | stride1 lo
    g1[7] = 0;
    int32x4 z4 = (int32x4){0, 0, 0, 0};
    TDM_CALL(g0, g1, z4, z4);
}
#define UPADBITS ((1u << 20) | (7u << 22) | (1u << 25)) // pad 2 dw every 256 dw
#endif // HAVE_TDM

__global__ __launch_bounds__(TW) void diffusion_step_kernel(
    const float* __restrict__ U, const float* __restrict__ A,
    const float* __restrict__ Bm, const float* __restrict__ Cm,
    const float* __restrict__ tau_p, const float* __restrict__ gradx,
    float* __restrict__ Out)
{
    __shared__ float UP[TH + 2][TW + 2];   // replicate-padded u tile (pitch 258)
    __shared__ float SA[PH][PW];           // a rows y0..y0+16, cols x0..x0+257
    __shared__ float SB[PH][PW];           // b  (same window)
    __shared__ float SC[PH][PW];           // c  (same window)

    const int tx    = threadIdx.x;
    const int wid   = tx >> 5;             // wave32 id (0..7)
    const int x0    = blockIdx.x * TW;
    const int y0    = blockIdx.y * TH;
    const int plane = blockIdx.z;

    const float* up = U  + (size_t)plane * (HH * WW);
    const float* Ap = A  + (size_t)plane * ((HH + 2) * (WW + 2));
    const float* Bp = Bm + (size_t)plane * ((HH + 2) * (WW + 2));
    const float* Cp = Cm + (size_t)plane * ((HH + 2) * (WW + 2));
    float*       op = Out + (size_t)plane * (HH * WW);

    const size_t pbase = (size_t)y0 * WPAD + x0;   // a/b/c tile origin (in-bounds)

#if HAVE_TDM
    const bool rows_interior = (y0 > 0) && (y0 + TH < HH);   // 62/64 row-blocks
    // Four concurrent DMAs, one per wave (EXEC all-ones per issuing wave,
    // blockIdx-uniform descriptors => SGPR operands).
    if (wid == 1) {
        tdm_load_2d(Ap + pbase, (unsigned)(uintptr_t)&SA[0][0], PH, PW, WPAD, 0u);
    } else if (wid == 2) {
        tdm_load_2d(Bp + pbase, (unsigned)(uintptr_t)&SB[0][0], PH, PW, WPAD, 0u);
    } else if (wid == 3) {
        tdm_load_2d(Cp + pbase, (unsigned)(uintptr_t)&SC[0][0], PH, PW, WPAD, 0u);
    } else if (wid == 0 && rows_interior) {
        tdm_load_2d(up + (size_t)(y0 - 1) * WW + x0,
                    (unsigned)(uintptr_t)&UP[0][1], TH + 2, TW, WW, UPADBITS);
    }
    if (rows_interior) {
        // u halo columns 0 and 257 (clamped in x) via normal loads, threads 0..35.
        if (tx < 2 * (TH + 2)) {
            int r    = tx % (TH + 2);
            int side = tx / (TH + 2);
            int gx   = side ? min(x0 + TW, WW - 1) : max(x0 - 1, 0);
            int gy   = y0 - 1 + r;                 // in-bounds (interior rows)
            UP[r][side ? (TW + 1) : 0] = up[(size_t)gy * WW + gx];
        }
    } else {
        // Boundary row-blocks: fully manual clamped u stage (TDM can't replicate).
        for (int i = tx; i < (TH + 2) * (TW + 2); i += TW) {
            int r  = i / (TW + 2);
            int c  = i - r * (TW + 2);
            int gy = min(max(y0 - 1 + r, 0), HH - 1);
            int gx = min(max(x0 - 1 + c, 0), WW - 1);
            UP[r][c] = up[(size_t)gy * WW + gx];
        }
    }
    __builtin_amdgcn_s_wait_tensorcnt(0);   // no-op for waves 4..7
#else
    // No-TDM toolchain: manual staging for everything.
    for (int i = tx; i < (TH + 2) * (TW + 2); i += TW) {
        int r  = i / (TW + 2);
        int c  = i - r * (TW + 2);
        int gy = min(max(y0 - 1 + r, 0), HH - 1);
        int gx = min(max(x0 - 1 + c, 0), WW - 1);
        UP[r][c] = up[(size_t)gy * WW + gx];
    }
    for (int i = tx; i < PH * PW; i += TW) {
        int r = i / PW, c = i - r * PW;
        size_t g = pbase + (size_t)r * WPAD + c;
        SA[r][c] = Ap[g];
        SB[r][c] = Bp[g];
        SC[r][c] = Cp[g];
    }
#endif
    __syncthreads();

    // ---- compute TH outputs per thread, rolling 3x3 u-window in registers ----
    const float hx  = gradx[5];        // = 1/1.4142 (grad_x[0,0,1,2])
    const float thx = tau_p[0] * hx;   // outer divergence scale
    const int   x   = x0 + tx;

    float r0c0 = UP[0][tx], r0c1 = UP[0][tx + 1], r0c2 = UP[0][tx + 2];
    float r1c0 = UP[1][tx], r1c1 = UP[1][tx + 1], r1c2 = UP[1][tx + 2];

#pragma unroll 4
    for (int ty = 0; ty < TH; ++ty) {
        const int y = y0 + ty;
        float r2c0 = UP[ty + 2][tx];
        float r2c1 = UP[ty + 2][tx + 1];
        float r2c2 = UP[ty + 2][tx + 2];

        // ux/uy at P1=(y,x), P0=(y,max(x-1,0)), P2=(max(y-1,0),x).
        float uxP1 = hx * (r1c2 - r1c1);
        float uyP1 = hx * (r2c1 - r1c1);
        float uxP0 = (x == 0) ? uxP1 : hx * (r1c1 - r1c0);  // pad-of-ux collapse
        float uyP0 = hx * (r2c0 - r1c0);                     // clamp-safe via tile
        float uxP2 = hx * (r0c2 - r0c1);                     // clamp-safe via tile
        float uyP2 = (y == 0) ? uyP1 : hx * (r1c1 - r0c1);  // pad-of-uy collapse

        // a/b/c from LDS: padded-grid (row y+1 | y, col x | x+1) -> local
        // (ty+1 | ty, tx | tx+1).
        float a0 = SA[ty + 1][tx];
        float a1 = SA[ty + 1][tx + 1];
        float b0 = SB[ty + 1][tx];
        float b1 = SB[ty + 1][tx + 1];
        float bp = SB[ty][tx + 1];
        float c1 = SC[ty + 1][tx + 1];
        float cp = SC[ty][tx + 1];

        float F1a = a0 * uxP0 + b0 * uyP0;    // F1[y+1, x]
        float F1b = a1 * uxP1 + b1 * uyP1;    // F1[y+1, x+1]
        float F2a = bp * uxP2 + cp * uyP2;    // F2[y,   x+1]
        float F2b = b1 * uxP1 + c1 * uyP1;    // F2[y+1, x+1]

        op[(size_t)y * WW + x] = r1c1 + thx * (F1a - F1b + F2a - F2b);

        r0c0 = r1c0; r0c1 = r1c1; r0c2 = r1c2;
        r1c0 = r2c0; r1c1 = r2c1; r1c2 = r2c2;
    }
}

extern "C" void kernel_launch(void* const* d_in, const int* in_sizes, int n_in,
                              void* d_out, int out_size, void* d_ws, size_t ws_size,
                              hipStream_t stream)
{
    (void)n_in; (void)out_size; (void)d_ws; (void)ws_size;
    const float* u   = (const float*)d_in[0];
    const float* a   = (const float*)d_in[1];
    const float* b   = (const float*)d_in[2];
    const float* c   = (const float*)d_in[3];
    const float* tau = (const float*)d_in[4];
    const float* gx  = (const float*)d_in[5];
    float* out       = (float*)d_out;

    const int planes = in_sizes[0] / (HH * WW);   // B*C = 16
    dim3 grid(WW / TW, HH / TH, planes);
    diffusion_step_kernel<<<grid, dim3(TW, 1, 1), 0, stream>>>(u, a, b, c, tau, gx, out);
}